// TriWindowMambaLayer_8615704396479
// MI455X (gfx1250) — compile-verified
//
#include <hip/hip_runtime.h>
#include <math.h>

#define DIMC 384
#define D_STATE 16
#define D_INNER 768
#define DT_RANK 24
#define LSEQ 4096
#define NZP 16
#define ZF 32

typedef float v2f __attribute__((ext_vector_type(2)));
typedef float v8f __attribute__((ext_vector_type(8)));

__device__ __forceinline__ float sigmoidf_(float x) { return 1.0f / (1.0f + __expf(-x)); }

// branch-local index -> original sequence index
__device__ __forceinline__ int map_idx(int l, int mode) {
    if (mode == 1) return LSEQ - 1 - l;                 // reversed branch
    if (mode == 2) return (l & 15) * 256 + (l >> 4);    // slice-interleave branch
    return l;                                           // forward branch
}

// ---------------------------------------------------------------------------
// Kernel 1: AvgPool3d(2) + LayerNorm over channels; writes xnT as (C, L)
// one block per pooled voxel l, 384 threads = one per channel
// ---------------------------------------------------------------------------
__global__ void pool_ln_kernel(const float* __restrict__ x,
                               const float* __restrict__ lnw,
                               const float* __restrict__ lnb,
                               float* __restrict__ xnT) {
    __shared__ float red[DIMC];
    int l = blockIdx.x;
    int c = threadIdx.x;
    int iz = l >> 8, ih = (l >> 4) & 15, iw = l & 15;
    const float* xc = x + (size_t)c * ZF * ZF * ZF;
    float v = 0.f;
#pragma unroll
    for (int a = 0; a < 2; a++)
#pragma unroll
        for (int b = 0; b < 2; b++)
#pragma unroll
            for (int d = 0; d < 2; d++)
                v += xc[((2 * iz + a) * ZF + (2 * ih + b)) * ZF + (2 * iw + d)];
    v *= 0.125f;

    red[c] = v;
    __syncthreads();
    for (int s = 192; s > 2; s >>= 1) { if (c < s) red[c] += red[c + s]; __syncthreads(); }
    float mu = (red[0] + red[1] + red[2]) * (1.0f / DIMC);
    __syncthreads();
    float dv = v - mu;
    red[c] = dv * dv;
    __syncthreads();
    for (int s = 192; s > 2; s >>= 1) { if (c < s) red[c] += red[c + s]; __syncthreads(); }
    float var = (red[0] + red[1] + red[2]) * (1.0f / DIMC);
    float rstd = rsqrtf(var + 1e-5f);
    xnT[(size_t)c * LSEQ + l] = dv * rstd * lnw[c] + lnb[c];
}

// ---------------------------------------------------------------------------
// Kernel 2: fp32 WMMA GEMM: Out[M x N] = W[M x K] * X[K x N]
// wave = 16(m) x 64(n) output tile -> 4 WMMAs per k-step sharing one A frag.
// All addresses strength-reduced to pointer increments; branch-free software
// prefetch on the streamed X operand (speculative, dropped if OOB).
// A frag: lane m = lane&15, half = lane>>4 selects k-pair.
// C/D:    vgpr i <-> row m0 + i + 8*half, col = ntile + (lane&15).
// ---------------------------------------------------------------------------
__global__ void gemm_wmma_f32(const float* __restrict__ Wp,
                              const float* __restrict__ Xp,
                              float* __restrict__ Outp,
                              int K, int N, int Mvalid) {
    int wave = threadIdx.x >> 5;
    int lane = threadIdx.x & 31;
    int n0 = (blockIdx.x * 4 + wave) * 64;   // 64 columns per wave
    int m0 = blockIdx.y * 16;
    int half = lane >> 4;
    int lidx = lane & 15;
    int ra = m0 + lidx; if (ra >= Mvalid) ra = Mvalid - 1;   // clamp padded rows
    int kA = half * 2;

    const float* ap = Wp + (size_t)ra * K + kA;          // A: k, k+1 (contiguous)
    const float* bp = Xp + (size_t)kA * N + n0 + lidx;   // B row k
    const float* bq = bp + N;                            // B row k+1
    const size_t bstep = (size_t)4 * N;
    const float* pf = bp + 8 * bstep;                    // prefetch 8 k-steps ahead

    v8f acc0, acc1, acc2, acc3;
#pragma unroll
    for (int i = 0; i < 8; i++) { acc0[i] = 0.f; acc1[i] = 0.f; acc2[i] = 0.f; acc3[i] = 0.f; }

    for (int k0 = 0; k0 < K; k0 += 4) {
        v2f a = *(const v2f*)ap;                         // 8B aligned (even offset)
        v2f b0, b1, b2, b3;
        b0.x = bp[0];  b0.y = bq[0];
        b1.x = bp[16]; b1.y = bq[16];
        b2.x = bp[32]; b2.y = bq[32];
        b3.x = bp[48]; b3.y = bq[48];
        __builtin_prefetch(pf, 0, 1);
        acc0 = __builtin_amdgcn_wmma_f32_16x16x4_f32(false, a, false, b0, (short)0, acc0, false, false);
        acc1 = __builtin_amdgcn_wmma_f32_16x16x4_f32(false, a, false, b1, (short)0, acc1, false, false);
        acc2 = __builtin_amdgcn_wmma_f32_16x16x4_f32(false, a, false, b2, (short)0, acc2, false, false);
        acc3 = __builtin_amdgcn_wmma_f32_16x16x4_f32(false, a, false, b3, (short)0, acc3, false, false);
        ap += 4; bp += bstep; bq += bstep; pf += bstep;
    }

    float* orow = Outp + (size_t)(m0 + half * 8) * N + n0 + lidx;
#pragma unroll
    for (int i = 0; i < 8; i++) {
        int rm = m0 + half * 8 + i;
        if (rm < Mvalid) {
            orow[0]  = acc0[i];
            orow[16] = acc1[i];
            orow[32] = acc2[i];
            orow[48] = acc3[i];
        }
        orow += N;
    }
}

// ---------------------------------------------------------------------------
// Kernel 3: causal depthwise conv (K=4) + SiLU, reading xz rows [0,768) with
// branch index mapping; writes xconv (768 x L) in branch-local order
// ---------------------------------------------------------------------------
__global__ void conv_silu_kernel(const float* __restrict__ xz,
                                 const float* __restrict__ cw,
                                 const float* __restrict__ cb,
                                 float* __restrict__ xconv,
                                 int mode) {
    int idx = blockIdx.x * blockDim.x + threadIdx.x;
    if (idx >= D_INNER * LSEQ) return;
    int d = idx >> 12;
    int l = idx & (LSEQ - 1);
    const float* row = xz + (size_t)d * LSEQ;
    float acc = cb[d];
#pragma unroll
    for (int k = 0; k < 4; k++) {
        int j = l - 3 + k;
        if (j >= 0) acc += cw[d * 4 + k] * row[map_idx(j, mode)];
    }
    xconv[idx] = acc * sigmoidf_(acc);
}

// ---------------------------------------------------------------------------
// Kernel 4: selective scan. 16 lanes = one channel's 16 states (h kept in a
// register), sequential over L; shfl-xor reduce h*C across the state group,
// fused softplus(dt)+bias, +D*u, *silu(z); writes osum at original index.
// block = 128 threads = 8 channels; grid = 96 blocks (768 channels)
// ---------------------------------------------------------------------------
__global__ void scan_kernel(const float* __restrict__ u,      // xconv (768 x L)
                            const float* __restrict__ xdbl,   // 64 x L (rows 24..39 B, 40..55 C)
                            const float* __restrict__ dpre,   // 768 x L (pre-bias dt projection)
                            const float* __restrict__ dtb,    // (768,)
                            const float* __restrict__ A_log,  // (768,16)
                            const float* __restrict__ Dp,     // (768,)
                            const float* __restrict__ xz,     // z from rows [768,1536)
                            float* __restrict__ osum,         // (768 x L) original order
                            int mode, int accum) {
    int grp = threadIdx.x >> 4;
    int n = threadIdx.x & 15;
    int d = blockIdx.x * 8 + grp;

    float A = -__expf(A_log[d * D_STATE + n]);
    float db = dtb[d];
    float Dd = Dp[d];
    const float* urow = u    + (size_t)d * LSEQ;
    const float* prow = dpre + (size_t)d * LSEQ;
    const float* Brow = xdbl + (size_t)(DT_RANK + n) * LSEQ;
    const float* Crow = xdbl + (size_t)(DT_RANK + D_STATE + n) * LSEQ;
    const float* zrow = xz   + (size_t)(D_INNER + d) * LSEQ;
    float* orow = osum + (size_t)d * LSEQ;

    float h = 0.f;
    for (int l = 0; l < LSEQ; l++) {
        float pre = prow[l] + db;
        float dlt = (pre > 20.f) ? pre : log1pf(__expf(pre));
        float uu = urow[l];
        h = __expf(dlt * A) * h + dlt * Brow[l] * uu;
        float p = h * Crow[l];
        p += __shfl_xor(p, 1, 16);
        p += __shfl_xor(p, 2, 16);
        p += __shfl_xor(p, 4, 16);
        p += __shfl_xor(p, 8, 16);
        if (n == 0) {
            int o = map_idx(l, mode);
            float zv = zrow[o];
            float yv = (p + Dd * uu) * zv * sigmoidf_(zv);
            if (accum) orow[o] += yv; else orow[o] = yv;
        }
    }
}

// ---------------------------------------------------------------------------
// Kernel 5: trilinear x2 upsample (align_corners=False) + residual add
// ---------------------------------------------------------------------------
__device__ __forceinline__ void lin_coef(int i, int n, int& i0, int& i1, float& w) {
    float src = (i + 0.5f) * 0.5f - 0.5f;
    src = fmaxf(src, 0.0f);
    i0 = (int)src;
    i1 = i0 + 1; if (i1 > n - 1) i1 = n - 1;
    w = src - (float)i0;
}

__global__ void upsample_add_kernel(const float* __restrict__ yT,  // (384 x 4096) = (C,16,16,16)
                                    const float* __restrict__ x,
                                    float* __restrict__ out) {
    int idx = blockIdx.x * blockDim.x + threadIdx.x;
    const int VOX = ZF * ZF * ZF;
    if (idx >= DIMC * VOX) return;
    int c = idx / VOX;
    int r = idx - c * VOX;
    int z = r >> 10, hh = (r >> 5) & 31, w = r & 31;
    int z0, z1, h0, h1, w0, w1; float wz, wh, ww;
    lin_coef(z, NZP, z0, z1, wz);
    lin_coef(hh, NZP, h0, h1, wh);
    lin_coef(w, NZP, w0, w1, ww);
    const float* yc = yT + (size_t)c * LSEQ;
    float v000 = yc[z0 * 256 + h0 * 16 + w0], v001 = yc[z0 * 256 + h0 * 16 + w1];
    float v010 = yc[z0 * 256 + h1 * 16 + w0], v011 = yc[z0 * 256 + h1 * 16 + w1];
    float v100 = yc[z1 * 256 + h0 * 16 + w0], v101 = yc[z1 * 256 + h0 * 16 + w1];
    float v110 = yc[z1 * 256 + h1 * 16 + w0], v111 = yc[z1 * 256 + h1 * 16 + w1];
    float a0 = v000 + ww * (v001 - v000);
    float a1 = v010 + ww * (v011 - v010);
    float b0 = v100 + ww * (v101 - v100);
    float b1 = v110 + ww * (v111 - v110);
    float c0 = a0 + wh * (a1 - a0);
    float c1 = b0 + wh * (b1 - b0);
    out[idx] = c0 + wz * (c1 - c0) + x[idx];
}

// ---------------------------------------------------------------------------
extern "C" void kernel_launch(void* const* d_in, const int* in_sizes, int n_in,
                              void* d_out, int out_size, void* d_ws, size_t ws_size,
                              hipStream_t stream) {
    const float* x    = (const float*)d_in[0];
    const float* lnw  = (const float*)d_in[1];
    const float* lnb  = (const float*)d_in[2];
    const float* ipw  = (const float*)d_in[3];   // (1536, 384)
    const float* opw  = (const float*)d_in[4];   // (384, 768)

    float* ws    = (float*)d_ws;
    float* xnT   = ws;                           // 384 x 4096
    float* xzb   = xnT  + (size_t)DIMC * LSEQ;   // 1536 x 4096
    float* xconv = xzb  + (size_t)2 * D_INNER * LSEQ;  // 768 x 4096
    float* xdbl  = xconv + (size_t)D_INNER * LSEQ;     // 64 x 4096 (56 valid)
    float* dpre  = xdbl + (size_t)64 * LSEQ;           // 768 x 4096
    float* osum  = dpre + (size_t)D_INNER * LSEQ;      // 768 x 4096
    float* yT    = osum + (size_t)D_INNER * LSEQ;      // 384 x 4096
    float* outp  = (float*)d_out;

    // 1) pool + layernorm -> xnT (C, L)
    pool_ln_kernel<<<LSEQ, DIMC, 0, stream>>>(x, lnw, lnb, xnT);

    // 2) in_proj: xz (1536 x 4096) = ipw (1536x384) * xnT (384x4096)
    gemm_wmma_f32<<<dim3(LSEQ / 256, 2 * D_INNER / 16), 128, 0, stream>>>(
        ipw, xnT, xzb, DIMC, LSEQ, 2 * D_INNER);

    // 3) three branches, sequential (forward / reversed / slice-interleaved)
    for (int br = 0; br < 3; br++) {
        int base = 5 + 7 * br;
        const float* cw  = (const float*)d_in[base + 0];  // (768,1,4)
        const float* cb  = (const float*)d_in[base + 1];  // (768,)
        const float* xw  = (const float*)d_in[base + 2];  // (56,768)
        const float* dw  = (const float*)d_in[base + 3];  // (768,24)
        const float* dbv = (const float*)d_in[base + 4];  // (768,)
        const float* Alg = (const float*)d_in[base + 5];  // (768,16)
        const float* Dv  = (const float*)d_in[base + 6];  // (768,)

        // depthwise causal conv + silu
        conv_silu_kernel<<<(D_INNER * LSEQ + 255) / 256, 256, 0, stream>>>(
            xzb, cw, cb, xconv, br);

        // xproj: xdbl (56 x 4096) = xw (56x768) * xconv (768x4096), M padded to 64
        gemm_wmma_f32<<<dim3(LSEQ / 256, 4), 128, 0, stream>>>(
            xw, xconv, xdbl, D_INNER, LSEQ, DT_RANK + 2 * D_STATE);

        // dtproj: dpre (768 x 4096) = dw (768x24) * xdbl[0:24,:] (24x4096)
        gemm_wmma_f32<<<dim3(LSEQ / 256, D_INNER / 16), 128, 0, stream>>>(
            dw, xdbl, dpre, DT_RANK, LSEQ, D_INNER);

        // selective scan + gate, accumulate into osum (original ordering)
        scan_kernel<<<D_INNER / 8, 128, 0, stream>>>(
            xconv, xdbl, dpre, dbv, Alg, Dv, xzb, osum, br, br != 0);
    }

    // 4) out_proj: yT (384 x 4096) = opw (384x768) * osum (768x4096)
    gemm_wmma_f32<<<dim3(LSEQ / 256, DIMC / 16), 128, 0, stream>>>(
        opw, osum, yT, D_INNER, LSEQ, DIMC);

    // 5) trilinear upsample x2 + residual
    upsample_add_kernel<<<(DIMC * ZF * ZF * ZF + 255) / 256, 256, 0, stream>>>(
        yT, x, outp);
}